// TemporalGNN_15341623181989
// MI455X (gfx1250) — compile-verified
//
#include <hip/hip_runtime.h>
#include <math.h>

typedef __attribute__((ext_vector_type(2))) float v2f;
typedef __attribute__((ext_vector_type(8))) float v8f;

#define Bb   128
#define Nn   2000
#define FIN  4
#define Pp   12
#define Hh   16
#define FOUT 12
#define Ee   32000
#define ROWS (Nn * Bb)        /* 256000, node-major rows: r = n*Bb + b */
#define TILES (ROWS / 16)     /* 16000 */

/* workspace layout (float elements) */
#define OFF_DEG   0ull
#define OFF_NORM  2048ull                                  /* needs Ee+Nn = 34000 */
#define OFF_PROBS 36864ull                                 /* needs 12 */
#define OFF_XW    37888ull                                 /* 3*ROWS*Hh = 12,288,000 */
#define OFF_GC    (OFF_XW + 3ull * ROWS * Hh)
#define OFF_HC    (OFF_GC + 3ull * ROWS * Hh)
#define OFF_HA    (OFF_HC + 1ull * ROWS * Hh)
/* total end = 32,805,888 floats ~= 131 MB */

__global__ void fill_kernel(float* p, float v, long long count) {
  long long i = (long long)blockIdx.x * blockDim.x + threadIdx.x;
  if (i < count) p[i] = v;
}

/* deg initialized to 1.0 (self-loop weight); accumulate edge weights on dst */
__global__ void deg_kernel(const int* __restrict__ ei, const float* __restrict__ ew,
                           float* __restrict__ deg) {
  int e = blockIdx.x * blockDim.x + threadIdx.x;
  if (e < Ee) atomicAdd(&deg[ei[Ee + e]], ew[e]);
}

/* norm for all E real edges + N self loops */
__global__ void norm_kernel(const int* __restrict__ ei, const float* __restrict__ ew,
                            const float* __restrict__ deg, float* __restrict__ nrm) {
  int e = blockIdx.x * blockDim.x + threadIdx.x;
  if (e >= Ee + Nn) return;
  int s, d; float w;
  if (e < Ee) { s = ei[e]; d = ei[Ee + e]; w = ew[e]; }
  else        { s = d = e - Ee; w = 1.0f; }
  float ds = deg[s], dd = deg[d];
  float is = ds > 0.f ? rsqrtf(ds) : 0.f;
  float id = dd > 0.f ? rsqrtf(dd) : 0.f;
  nrm[e] = is * w * id;
}

__global__ void softmax_kernel(const float* __restrict__ att, float* __restrict__ probs) {
  if (threadIdx.x == 0) {
    float m = -1e30f;
    for (int i = 0; i < Pp; ++i) m = fmaxf(m, att[i]);
    float ex[Pp]; float s = 0.f;
    for (int i = 0; i < Pp; ++i) { ex[i] = __expf(att[i] - m); s += ex[i]; }
    for (int i = 0; i < Pp; ++i) probs[i] = ex[i] / s;
  }
}

/* XW for all 3 gates at time step p: (ROWS x 4) @ (4 x 16), one f32 WMMA per tile.
   Output layout: xw[gate][r = n*Bb + b][h], so each node owns a contiguous Bb*Hh block. */
__global__ void __launch_bounds__(256) xw_kernel(const float* __restrict__ x,
                                                 const float* __restrict__ Wz,
                                                 const float* __restrict__ Wr,
                                                 const float* __restrict__ Wh,
                                                 float* __restrict__ xw, int p) {
  const int lane = threadIdx.x & 31;
  const int wave = threadIdx.x >> 5;
  const int tile = blockIdx.x * 8 + wave;     /* grid.x = TILES/8 exactly */
  const int gate = blockIdx.y;
  const float* W = (gate == 0) ? Wz : ((gate == 1) ? Wr : Wh);

  const int arow = lane & 15;                 /* A-matrix M index */
  const int hi   = lane >> 4;
  const int k0   = hi * 2;                    /* K halves per 16x4 f32 A layout */
  const int r    = tile * 16 + arow;
  const int n    = r >> 7;                    /* r / Bb, Bb == 128 */
  const int b    = r & 127;

  const float* xp = x + ((((long long)b * Nn + n) * FIN + k0) * Pp + p);
  v2f a;  a.x = xp[0];           a.y = xp[Pp];            /* f stride is Pp */
  v2f bm; bm.x = W[k0 * Hh + arow]; bm.y = W[(k0 + 1) * Hh + arow];
  v8f c = {};
  c = __builtin_amdgcn_wmma_f32_16x16x4_f32(false, a, false, bm, (short)0, c, false, false);

  float* outp = xw + ((long long)gate * ROWS + (long long)tile * 16) * Hh;
  const int col = lane & 15;
#pragma unroll
  for (int i = 0; i < 8; ++i)
    outp[(hi * 8 + i) * Hh + col] = c[i];
}

/* gc[gate][dst-block] += norm * xw[gate][src-block]; one block per (edge, gate),
   contiguous 2048-float node blocks -> fully coalesced atomics. */
__global__ void scatter_kernel(const int* __restrict__ ei, const float* __restrict__ nrm,
                               const float* __restrict__ xw, float* __restrict__ gc) {
  const int e = blockIdx.x;
  const int gate = blockIdx.y;
  int s, d;
  if (e < Ee) { s = ei[e]; d = ei[Ee + e]; }
  else        { s = d = e - Ee; }
  const float w = nrm[e];
  const float* sp = xw + (long long)gate * ROWS * Hh + (long long)s * Bb * Hh;
  float*       dp = gc + (long long)gate * ROWS * Hh + (long long)d * Bb * Hh;
  for (int t = threadIdx.x; t < Bb * Hh; t += blockDim.x)
    atomicAdd(&dp[t], w * sp[t]);
}

/* One wave = one 16-row tile. Per gate: A = [gc_g + bias_g | Hc (or Hc*R)] (16x32),
   B = gate weight transposed (32x16), C init = output bias -> 8-chain of f32 WMMAs. */
__global__ void __launch_bounds__(256) gru_kernel(
    const float* __restrict__ gc,
    const float* __restrict__ bz, const float* __restrict__ br, const float* __restrict__ bh,
    const float* __restrict__ lzw, const float* __restrict__ lzb,
    const float* __restrict__ lrw, const float* __restrict__ lrb,
    const float* __restrict__ lhw, const float* __restrict__ lhb,
    const float* __restrict__ probs, int p,
    float* __restrict__ Hc, float* __restrict__ Ha) {
  __shared__ float s_w[3][32][16];   /* B matrices: s_w[g][k][n] = l?w[n*32+k] */
  __shared__ float s_cb[3][16];      /* output biases lzb/lrb/lhb */
  __shared__ float s_gb[3][16];      /* gconv biases bz/br/bh */
  __shared__ float s_a[8][16][32];   /* per-wave A staging */
  __shared__ float s_hc[8][16][16];  /* per-wave Hc tile */
  __shared__ float s_r[8][16][16];   /* per-wave R tile */

  const int tid = threadIdx.x;
  for (int i = tid; i < 512; i += 256) {
    int nn = i >> 5, kk = i & 31;
    s_w[0][kk][nn] = lzw[i];
    s_w[1][kk][nn] = lrw[i];
    s_w[2][kk][nn] = lhw[i];
  }
  if (tid < 16) {
    s_cb[0][tid] = lzb[tid]; s_cb[1][tid] = lrb[tid]; s_cb[2][tid] = lhb[tid];
    s_gb[0][tid] = bz[tid];  s_gb[1][tid] = br[tid];  s_gb[2][tid] = bh[tid];
  }
  __syncthreads();

  const int lane = tid & 31;
  const int wave = tid >> 5;
  const int tile = blockIdx.x * 8 + wave;    /* grid.x = TILES/8 exactly */
  const int row0 = tile * 16;
  const int col  = lane & 15;
  const int hi   = lane >> 4;
  const float pw = probs[p];

  /* stage Hc tile (in-wave LDS ordering keeps this coherent for this wave) */
  for (int i = lane; i < 256; i += 32) {
    int rr = i >> 4, cc = i & 15;
    s_hc[wave][rr][cc] = Hc[(long long)(row0 + rr) * Hh + cc];
  }

  v8f frag[3];
#pragma unroll 1
  for (int g = 0; g < 3; ++g) {
    const float* gcg = gc + (long long)g * ROWS * Hh;
    for (int i = lane; i < 512; i += 32) {
      int rr = i >> 5, cc = i & 31;
      float v;
      if (cc < 16) {
        v = gcg[(long long)(row0 + rr) * Hh + cc] + s_gb[g][cc];
      } else {
        v = s_hc[wave][rr][cc - 16];
        if (g == 2) v *= s_r[wave][rr][cc - 16];
      }
      s_a[wave][rr][cc] = v;
    }

    v8f c;
#pragma unroll
    for (int i = 0; i < 8; ++i) c[i] = s_cb[g][col];

#pragma unroll
    for (int kk = 0; kk < 8; ++kk) {
      const int k0 = kk * 4 + hi * 2;
      v2f a;  a.x  = s_a[wave][col][k0];  a.y  = s_a[wave][col][k0 + 1];
      v2f bm; bm.x = s_w[g][k0][col];     bm.y = s_w[g][k0 + 1][col];
      c = __builtin_amdgcn_wmma_f32_16x16x4_f32(false, a, false, bm, (short)0, c, false, false);
    }

    if (g < 2) {
#pragma unroll
      for (int i = 0; i < 8; ++i) c[i] = 1.0f / (1.0f + __expf(-c[i]));
      if (g == 1) {
#pragma unroll
        for (int i = 0; i < 8; ++i) s_r[wave][hi * 8 + i][col] = c[i];
      }
    } else {
#pragma unroll
      for (int i = 0; i < 8; ++i) c[i] = tanhf(c[i]);
    }
    frag[g] = c;
  }

  /* Hn = Z*Hc + (1-Z)*Ht ; update state and weighted accumulator */
#pragma unroll
  for (int i = 0; i < 8; ++i) {
    const int rr = hi * 8 + i;
    const float hc = s_hc[wave][rr][col];
    const float z  = frag[0][i];
    const float hn = z * hc + (1.0f - z) * frag[2][i];
    const long long idx = (long long)(row0 + rr) * Hh + col;
    Hc[idx] = hn;
    Ha[idx] += pw * hn;
  }
}

/* out[b][n][f] = relu(Ha[n*Bb+b][:]) @ lin_w[f][:] + lin_b[f] */
__global__ void out_kernel(const float* __restrict__ Ha,
                           const float* __restrict__ lin_w, const float* __restrict__ lin_b,
                           float* __restrict__ out) {
  long long i = (long long)blockIdx.x * blockDim.x + threadIdx.x;
  if (i >= (long long)Bb * Nn * FOUT) return;
  int f = (int)(i % FOUT);
  long long bn = i / FOUT;
  int n = (int)(bn % Nn);
  int b = (int)(bn / Nn);
  const float* hp = Ha + ((long long)n * Bb + b) * Hh;
  float acc = lin_b[f];
#pragma unroll
  for (int h = 0; h < Hh; ++h) acc += fmaxf(hp[h], 0.f) * lin_w[f * Hh + h];
  out[i] = acc;
}

extern "C" void kernel_launch(void* const* d_in, const int* in_sizes, int n_in,
                              void* d_out, int out_size, void* d_ws, size_t ws_size,
                              hipStream_t stream) {
  const float* x   = (const float*)d_in[0];
  const float* ew  = (const float*)d_in[1];
  const float* Wz  = (const float*)d_in[2];
  const float* bz  = (const float*)d_in[3];
  const float* lzw = (const float*)d_in[4];
  const float* lzb = (const float*)d_in[5];
  const float* Wr  = (const float*)d_in[6];
  const float* br  = (const float*)d_in[7];
  const float* lrw = (const float*)d_in[8];
  const float* lrb = (const float*)d_in[9];
  const float* Wh  = (const float*)d_in[10];
  const float* bh  = (const float*)d_in[11];
  const float* lhw = (const float*)d_in[12];
  const float* lhb = (const float*)d_in[13];
  const float* att = (const float*)d_in[14];
  const float* lw  = (const float*)d_in[15];
  const float* lb  = (const float*)d_in[16];
  const int*   ei  = (const int*)d_in[17];

  float* ws   = (float*)d_ws;
  float* deg  = ws + OFF_DEG;
  float* nrm  = ws + OFF_NORM;
  float* prb  = ws + OFF_PROBS;
  float* xw   = ws + OFF_XW;
  float* gcb  = ws + OFF_GC;
  float* Hc   = ws + OFF_HC;
  float* Ha   = ws + OFF_HA;

  const long long stateN = (long long)ROWS * Hh;       /* 4,096,000 */

  fill_kernel<<<(Nn + 255) / 256, 256, 0, stream>>>(deg, 1.0f, Nn);
  fill_kernel<<<(int)((stateN + 255) / 256), 256, 0, stream>>>(Hc, 0.f, stateN);
  fill_kernel<<<(int)((stateN + 255) / 256), 256, 0, stream>>>(Ha, 0.f, stateN);
  deg_kernel<<<(Ee + 255) / 256, 256, 0, stream>>>(ei, ew, deg);
  norm_kernel<<<(Ee + Nn + 255) / 256, 256, 0, stream>>>(ei, ew, deg, nrm);
  softmax_kernel<<<1, 32, 0, stream>>>(att, prb);

  for (int p = 0; p < Pp; ++p) {
    xw_kernel<<<dim3(TILES / 8, 3), 256, 0, stream>>>(x, Wz, Wr, Wh, xw, p);
    fill_kernel<<<(int)((3 * stateN + 255) / 256), 256, 0, stream>>>(gcb, 0.f, 3 * stateN);
    scatter_kernel<<<dim3(Ee + Nn, 3), 256, 0, stream>>>(ei, nrm, xw, gcb);
    gru_kernel<<<TILES / 8, 256, 0, stream>>>(gcb, bz, br, bh, lzw, lzb, lrw, lrb,
                                              lhw, lhb, prb, p, Hc, Ha);
  }

  out_kernel<<<(int)(((long long)Bb * Nn * FOUT + 255) / 256), 256, 0, stream>>>(
      Ha, lw, lb, (float*)d_out);
}